// RegionLayer_38809324487148
// MI455X (gfx1250) — compile-verified
//
#include <hip/hip_runtime.h>
#include <hip/hip_bf16.h>

#define NBATCH 16
#define NANCH  5
#define NCLS   80
#define NHH    64
#define NWW    64
#define MAXT   50
#define NPIX   (NHH * NWW)          /* 4096  */
#define CPB    (NANCH * NPIX)       /* cells per batch: 20480 */
#define NTOTAL (NBATCH * CPB)       /* 327680 */
#define NCHAN  (NANCH * (5 + NCLS)) /* 425 */
#define RECF   12                   /* floats per target record (48 B, 16B-aligned) */
#define REC_OFF 32                  /* float offset of records in ws */
#define NV_OFF  8                   /* int offset of nvalid[] in ws */
#define THRESH_ 0.6f
#define OBJ_SCALE_ 5.0f

typedef unsigned int uint32;
typedef uint32 u32x4 __attribute__((ext_vector_type(4)));
typedef int    i32x4 __attribute__((ext_vector_type(4)));
typedef int    i32x8 __attribute__((ext_vector_type(8)));
typedef float  v2f   __attribute__((ext_vector_type(2)));
typedef float  v4f   __attribute__((ext_vector_type(4)));
typedef float  v8f   __attribute__((ext_vector_type(8)));

__device__ __forceinline__ float fast_rcp(float x) {
    return __builtin_amdgcn_rcpf(x);          // v_rcp_f32 (TRANS pipe)
}
__device__ __forceinline__ float fast_sigmoid(float x) {
    return fast_rcp(1.0f + __expf(-x));
}

// ---------------------------------------------------------------------------
// Setup: precompute per-target records + valid counts, zero loss accumulator.
// Record layout (12 floats): xlo,xhi,ylo,yhi, area, cell(int bits),
//                            t0,t1,t2,t3, gcls(int bits), pad
// ---------------------------------------------------------------------------
__global__ void region_setup(const float* __restrict__ target,
                             const float* __restrict__ anchors,
                             float* __restrict__ ws) {
    const int tid = threadIdx.x;
    __shared__ int s_nv[NBATCH];
    if (tid < NBATCH) {
        int nv = 0;
        for (int t = 0; t < MAXT; ++t) {
            if (target[tid * MAXT * 5 + t * 5 + 1] != 0.0f) nv++;
            else break;                       // cumprod semantics
        }
        s_nv[tid] = nv;
        ((int*)ws)[NV_OFF + tid] = nv;
        if (tid == 0) ws[0] = 0.0f;           // loss accumulator
    }
    __syncthreads();
    for (int idx = tid; idx < NBATCH * MAXT; idx += blockDim.x) {
        const int b = idx / MAXT, t = idx % MAXT;
        float* rec = ws + REC_OFF + idx * RECF;
        if (t >= s_nv[b]) {
            #pragma unroll
            for (int k = 0; k < RECF; ++k) rec[k] = 0.0f;
            rec[5] = __int_as_float(-1);
            continue;
        }
        const float* tb = target + b * MAXT * 5 + t * 5;
        const float gcls = tb[0];
        const float gx = tb[1] * (float)NWW, gy = tb[2] * (float)NHH;
        const float gw = tb[3] * (float)NWW, gh = tb[4] * (float)NHH;
        // best anchor (argmax of aligned IoU; first max wins like jnp.argmax)
        int   bn = 0;
        float best = -1.0f;
        #pragma unroll
        for (int a = 0; a < NANCH; ++a) {
            const float aw = anchors[a * 2], ah = anchors[a * 2 + 1];
            const float inter = fminf(gw, aw) * fminf(gh, ah);
            const float uni   = gw * gh + aw * ah - inter;
            const float r     = inter * fast_rcp(uni);
            if (r > best) { best = r; bn = a; }
        }
        const int gi = min(max((int)gx, 0), NWW - 1);
        const int gj = min(max((int)gy, 0), NHH - 1);
        rec[0]  = gx - 0.5f * gw;
        rec[1]  = gx + 0.5f * gw;
        rec[2]  = gy - 0.5f * gh;
        rec[3]  = gy + 0.5f * gh;
        rec[4]  = gw * gh;
        rec[5]  = __int_as_float(bn * NPIX + gj * NWW + gi);
        rec[6]  = gx - (float)gi;
        rec[7]  = gy - (float)gj;
        rec[8]  = __logf(gw * fast_rcp(anchors[bn * 2]));
        rec[9]  = __logf(gh * fast_rcp(anchors[bn * 2 + 1]));
        rec[10] = __int_as_float((int)gcls);
        rec[11] = 0.0f;
    }
}

// ---------------------------------------------------------------------------
// TDM: async 1-D dword copy global -> LDS (gfx1250 Tensor Data Mover)
// ---------------------------------------------------------------------------
__device__ __forceinline__ void tdm_load_dwords(uint32 lds_byte_addr,
                                                const float* gsrc,
                                                int ndwords) {
    const unsigned long long ga = (unsigned long long)(uintptr_t)gsrc;
    u32x4 g0;
    g0[0] = 1u;                                            // count=1, user mode
    g0[1] = lds_byte_addr;                                 // lds_addr
    g0[2] = (uint32)(ga & 0xFFFFFFFFull);                  // global_addr[31:0]
    g0[3] = (uint32)((ga >> 32) & 0x01FFFFFFull) | (2u << 30); // [56:32] | type=2
    i32x8 g1;
    g1[0] = (2 << 16);                                     // data_size = 4B
    g1[1] = (ndwords & 0xFFFF) << 16;                      // tensor_dim0 lo16
    g1[2] = ((ndwords >> 16) & 0xFFFF) | (1 << 16);        // td0 hi16 | td1 lo16=1
    g1[3] = (ndwords & 0xFFFF) << 16;                      // tile_dim0
    g1[4] = 0;                                             // tile_dim1/2 unused
    g1[5] = ndwords;                                       // td0_stride lo32
    g1[6] = 0;
    g1[7] = 0;
    i32x4 gz = {0, 0, 0, 0};
#if defined(__clang_major__) && (__clang_major__ >= 23)
    i32x8 gz8 = {0, 0, 0, 0, 0, 0, 0, 0};
    __builtin_amdgcn_tensor_load_to_lds(g0, g1, gz, gz, gz8, 0);
#else
    __builtin_amdgcn_tensor_load_to_lds(g0, g1, gz, gz, 0);
#endif
}

// ---------------------------------------------------------------------------
// Wave32 sum via v_wmma_f32_16x16x4_f32 with B = ones:
// D[m][n] = p_m + p_{m+16}; lane n<16 rows cover m=0..7, lane n+16 m=8..15.
// ---------------------------------------------------------------------------
__device__ __forceinline__ float wave_sum_wmma(float p) {
    v2f a; a[0] = p;    a[1] = 0.0f;
    v2f b; b[0] = 1.0f; b[1] = 1.0f;
    v8f c = {0.f, 0.f, 0.f, 0.f, 0.f, 0.f, 0.f, 0.f};
    c = __builtin_amdgcn_wmma_f32_16x16x4_f32(false, a, false, b,
                                              (short)0, c, false, false);
    return c[0] + c[1] + c[2] + c[3] + c[4] + c[5] + c[6] + c[7];
}

// ---------------------------------------------------------------------------
// Main: one thread per cell; one batch per 256-thread block.
// ---------------------------------------------------------------------------
__global__ __launch_bounds__(256) void region_main(
        const float* __restrict__ out,
        const float* __restrict__ anchors,
        float* __restrict__ ws) {
    __shared__ float srec[MAXT * RECF];   // 600 floats = 2400 B, 16B-aligned recs
    __shared__ float sred[16];

    const int tid = threadIdx.x;
    const int g   = blockIdx.x * 256 + tid;
    const int b   = g / CPB;
    const int rem = g % CPB;              // a*NPIX + j*NW + i  (== cell id)
    const int a   = rem / NPIX;
    const int pix = rem % NPIX;
    const int j   = pix >> 6;
    const int i   = pix & 63;

    // Stage this batch's target records into LDS via the TDM (wave 0 only),
    // overlapped with the channel loads + transcendental precompute below.
    if (tid < 32) {
        tdm_load_dwords((uint32)(uintptr_t)srec,
                        ws + REC_OFF + (size_t)b * (MAXT * RECF),
                        MAXT * RECF);
    }

    const int   nvalid = ((const int*)ws)[NV_OFF + b];
    const float aw = anchors[a * 2], ah = anchors[a * 2 + 1];

    const int base = (b * NCHAN + a * 85) * NPIX + pix;   // coalesced over i
    const float o0 = out[base];
    const float o1 = out[base + NPIX];
    const float o2 = out[base + 2 * NPIX];
    const float o3 = out[base + 3 * NPIX];
    const float o4 = out[base + 4 * NPIX];

    const float tx   = fast_sigmoid(o0);
    const float ty   = fast_sigmoid(o1);
    const float conf = fast_sigmoid(o4);
    const float pw   = __expf(o2) * aw;
    const float ph   = __expf(o3) * ah;
    const float px   = tx + (float)i;
    const float py   = ty + (float)j;
    const float pxl  = px - 0.5f * pw, pxh = px + 0.5f * pw;
    const float pyl  = py - 0.5f * ph, pyh = py + 0.5f * ph;
    const float parea = pw * ph;

    if (tid < 32) { __builtin_amdgcn_s_wait_tensorcnt(0); }
    __syncthreads();

    // IoU scan over valid targets; simultaneously find my matched target
    // (last writer wins, matching the JAX scatter).
    float maxiou = 0.0f, miou = 0.0f;
    int   match  = -1;
    for (int t = 0; t < nvalid; ++t) {
        const v4f box = *(const v4f*)(srec + t * RECF);      // ds_load_b128
        const v2f ac  = *(const v2f*)(srec + t * RECF + 4);  // ds_load_b64
        const float cw    = fminf(box[1], pxh) - fmaxf(box[0], pxl);
        const float chh   = fminf(box[3], pyh) - fmaxf(box[2], pyl);
        const float inter = fmaxf(cw, 0.0f) * fmaxf(chh, 0.0f);
        const float uni   = ac[0] + parea - inter;
        const float iou   = (inter > 0.0f) ? inter * fast_rcp(uni) : 0.0f;
        maxiou = fmaxf(maxiou, iou);
        if (__float_as_int(ac[1]) == rem) { match = t; miou = iou; }
    }

    float t0 = 0.5f, t1 = 0.5f, t2 = 0.0f, t3 = 0.0f;
    float cmask = (maxiou > THRESH_) ? 0.0f : 1.0f;
    float tconf = 0.0f;
    float ce    = 0.0f;
    if (match >= 0) {                      // <= ~20 lanes per batch take this
        const float* r = srec + match * RECF;
        t0 = r[6]; t1 = r[7]; t2 = r[8]; t3 = r[9];
        cmask = OBJ_SCALE_;
        tconf = miou;
        const int cls = __float_as_int(r[10]);
        // online log-sum-exp over the 80 class logits of this cell
        const float* lg = out + base + 5 * NPIX;
        float m = -INFINITY, s = 0.0f, sel = 0.0f;
        for (int c = 0; c < NCLS; ++c) {
            const float x  = lg[c * NPIX];
            const float mn = fmaxf(m, x);
            s = s * __expf(m - mn) + __expf(x - mn);
            m = mn;
            if (c == cls) sel = x;
        }
        ce = (m + __logf(s)) - sel;        // cross-entropy for matched cell
    }

    const float d0 = tx - t0, d1 = ty - t1, d2 = o2 - t2, d3 = o3 - t3;
    const float dc = conf - tconf;
    const float p  = 0.5f * (d0 * d0 + d1 * d1 + d2 * d2 + d3 * d3)
                   + 0.5f * cmask * dc * dc + ce;

    // wave32 reduction on the matrix pipe, then LDS + one atomic per block
    const float s8   = wave_sum_wmma(p);
    const int   lane = tid & 31, wid = tid >> 5;
    if ((lane & 15) == 0) sred[wid * 2 + (lane >> 4)] = s8;
    __syncthreads();
    if (tid == 0) {
        float tot = 0.0f;
        #pragma unroll
        for (int k = 0; k < 16; ++k) tot += sred[k];
        atomicAdd(ws, tot);
    }
}

__global__ void region_finalize(const float* __restrict__ ws,
                                float* __restrict__ out) {
    out[0] = ws[0];
}

extern "C" void kernel_launch(void* const* d_in, const int* in_sizes, int n_in,
                              void* d_out, int out_size, void* d_ws, size_t ws_size,
                              hipStream_t stream) {
    const float* output  = (const float*)d_in[0];
    const float* target  = (const float*)d_in[1];
    const float* anchors = (const float*)d_in[2];
    float* ws = (float*)d_ws;

    region_setup<<<1, 256, 0, stream>>>(target, anchors, ws);
    region_main<<<NTOTAL / 256, 256, 0, stream>>>(output, anchors, ws);
    region_finalize<<<1, 1, 0, stream>>>(ws, (float*)d_out);
}